// GAT_GNN_Edge_Feats_35579509080108
// MI455X (gfx1250) — compile-verified
//
#include <hip/hip_runtime.h>
#include <hip/hip_bf16.h>

// ---------------------------------------------------------------------------
// GAT GNN w/ edge feats for gfx1250 (MI455X, wave32, WMMA).
// Key optimization: (edge_attr @ We) . a_e  ==  edge_attr @ (We @ a_e),
// so all six 400000x768x256 edge GEMMs collapse to one 400000x768x6 pass.
// Node GEMMs use V_WMMA_F32_16X16X4_F32 with a 64KB LDS-staged B panel,
// K-pair interleaved so each WMMA fragment is a single ds_load_b64.
// ---------------------------------------------------------------------------

typedef __attribute__((ext_vector_type(2))) float v2f;
typedef __attribute__((ext_vector_type(8))) float v8f;

#define NNODES 50000
#define NEDGES 400000
#define HID    256
#define EDIM   768
#define NLAY   6

// ----- monotonic float<->uint key for atomic float max -----
__device__ __forceinline__ unsigned fkey(float f) {
    unsigned u = __float_as_uint(f);
    return (u & 0x80000000u) ? ~u : (u | 0x80000000u);
}
__device__ __forceinline__ float funkey(unsigned k) {
    unsigned u = (k & 0x80000000u) ? (k & 0x7FFFFFFFu) : ~k;
    return __uint_as_float(u);
}

// ---------------------------------------------------------------------------
// C[M,256] = op(A)[M,256] @ B[256,256], row-major.  op = relu if RELU.
// Block: 256 threads = 8 waves.  Block tile: 128(M) x 64(N), grid.y = 4.
// Each wave: 16(M) x 64(N) via 4 f32 WMMA accumulators, K stepped by 4.
// B panel staged in LDS K-pair interleaved: Bs[(k>>1)*128 + n*2 + (k&1)],
// so every B fragment (K-pair x same N) is one aligned 8-byte LDS load.
// ---------------------------------------------------------------------------
template <int RELU>
__global__ void gemm256_wmma_f32(const float* __restrict__ A,
                                 const float* __restrict__ B,
                                 float* __restrict__ C,
                                 int M) {
    __shared__ float Bs[256 * 64];           // 64 KB of the 320 KB WGP LDS
    const int tid = threadIdx.x;
    const int n0  = blockIdx.y * 64;

    // cooperative load of the 256x64 B panel, K-pair interleaved
    for (int j = tid; j < 256 * 64; j += 256) {
        int pair = j >> 7;                   // k >> 1
        int rem  = j & 127;
        int nl   = rem >> 1;
        int par  = rem & 1;
        Bs[j] = B[(pair * 2 + par) * 256 + n0 + nl];
    }
    __syncthreads();

    const int wave = tid >> 5;
    const int lane = tid & 31;
    const int row0 = (blockIdx.x * 8 + wave) * 16;
    if (row0 >= M) return;               // wave-uniform: EXEC stays all-ones

    const int mr = lane & 15;            // A: row within tile
    const int nc = lane & 15;            // B/C: col within tile
    const int kb = (lane >> 4) << 1;     // K-pair select (lanes 16-31 -> K+2)
    const int hi = lane >> 4;

    v8f acc0 = {}, acc1 = {}, acc2 = {}, acc3 = {};
    const float* Arow = A + (size_t)(row0 + mr) * HID;

    #pragma unroll 4
    for (int k0 = 0; k0 < HID; k0 += 4) {
        // A fragment: one aligned 8B global load (k0+kb is even)
        v2f a = *(const v2f*)(Arow + k0 + kb);
        if (RELU) {
            a.x = a.x > 0.f ? a.x : 0.f;
            a.y = a.y > 0.f ? a.y : 0.f;
        }

        // B fragments: one aligned 8B LDS load each
        const float* bp = &Bs[((k0 + kb) >> 1) * 128];
        v2f b0 = *(const v2f*)(bp + nc * 2);
        v2f b1 = *(const v2f*)(bp + (16 + nc) * 2);
        v2f b2 = *(const v2f*)(bp + (32 + nc) * 2);
        v2f b3 = *(const v2f*)(bp + (48 + nc) * 2);

        acc0 = __builtin_amdgcn_wmma_f32_16x16x4_f32(false, a, false, b0, (short)0, acc0, false, false);
        acc1 = __builtin_amdgcn_wmma_f32_16x16x4_f32(false, a, false, b1, (short)0, acc1, false, false);
        acc2 = __builtin_amdgcn_wmma_f32_16x16x4_f32(false, a, false, b2, (short)0, acc2, false, false);
        acc3 = __builtin_amdgcn_wmma_f32_16x16x4_f32(false, a, false, b3, (short)0, acc3, false, false);
    }

    // C/D layout: VGPR r -> row r + 8*hi, lane&15 -> col
    #pragma unroll
    for (int r = 0; r < 8; ++r) {
        float* crow = C + (size_t)(row0 + r + 8 * hi) * HID + n0 + nc;
        crow[0]  = acc0[r];
        crow[16] = acc1[r];
        crow[32] = acc2[r];
        crow[48] = acc3[r];
    }
}

// ---------------------------------------------------------------------------
// Vt[i][k] = sum_c We[i][k][c] * att_edge[i][c]   (6 x 768)
// ---------------------------------------------------------------------------
__global__ void ve_kernel(const float* __restrict__ We,
                          const float* __restrict__ att_edge,
                          float* __restrict__ Vt) {
    int idx = blockIdx.x * blockDim.x + threadIdx.x;
    if (idx >= NLAY * EDIM) return;
    int i = idx / EDIM, k = idx % EDIM;
    const float* w = We + ((size_t)i * EDIM + k) * HID;
    const float* a = att_edge + (size_t)i * HID;
    float s = 0.f;
    #pragma unroll 8
    for (int c = 0; c < HID; ++c) s += w[c] * a[c];
    Vt[(size_t)i * EDIM + k] = s;
}

// ---------------------------------------------------------------------------
// elog[e][i] = edge_attr[e] . Vt[i]   -- one streaming pass over edge_attr.
// One wave per edge, 8 edges per block; Vt (18KB) staged in LDS.
// ---------------------------------------------------------------------------
__global__ void edge_logits_kernel(const float* __restrict__ EA,
                                   const float* __restrict__ Vt,
                                   float* __restrict__ elog, int E) {
    __shared__ float Vs[NLAY * EDIM];
    for (int i = threadIdx.x; i < NLAY * EDIM; i += blockDim.x) Vs[i] = Vt[i];
    __syncthreads();

    const int wave = threadIdx.x >> 5;
    const int lane = threadIdx.x & 31;
    const int e = blockIdx.x * 8 + wave;
    if (e >= E) return;

    const float* row = EA + (size_t)e * EDIM;
    float acc[NLAY] = {0.f, 0.f, 0.f, 0.f, 0.f, 0.f};
    #pragma unroll
    for (int it = 0; it < 6; ++it) {                 // 6 * 128 = 768
        int base = it * 128 + lane * 4;
        float4 x = *(const float4*)(row + base);
        #pragma unroll
        for (int i = 0; i < NLAY; ++i) {
            const float* v = &Vs[i * EDIM + base];
            acc[i] += x.x * v[0] + x.y * v[1] + x.z * v[2] + x.w * v[3];
        }
    }
    #pragma unroll
    for (int i = 0; i < NLAY; ++i)
        for (int off = 16; off; off >>= 1) acc[i] += __shfl_xor(acc[i], off, 32);
    if (lane == 0) {
        #pragma unroll
        for (int i = 0; i < NLAY; ++i) elog[(size_t)e * NLAY + i] = acc[i];
    }
}

// ---------------------------------------------------------------------------
// s1[n] = hW[n].att_src, s2[n] = hW[n].att_dst   (one wave per node)
// ---------------------------------------------------------------------------
__global__ void node_scores_kernel(const float* __restrict__ hW,
                                   const float* __restrict__ as_,
                                   const float* __restrict__ ad_,
                                   float* __restrict__ s1,
                                   float* __restrict__ s2, int N) {
    const int wave = threadIdx.x >> 5;
    const int lane = threadIdx.x & 31;
    const int n = blockIdx.x * 8 + wave;
    if (n >= N) return;
    const float* row = hW + (size_t)n * HID;
    float a1 = 0.f, a2 = 0.f;
    #pragma unroll
    for (int it = 0; it < 2; ++it) {
        int base = it * 128 + lane * 4;
        float4 x = *(const float4*)(row + base);
        float4 u = *(const float4*)(as_ + base);
        float4 v = *(const float4*)(ad_ + base);
        a1 += x.x * u.x + x.y * u.y + x.z * u.z + x.w * u.w;
        a2 += x.x * v.x + x.y * v.y + x.z * v.z + x.w * v.w;
    }
    for (int off = 16; off; off >>= 1) {
        a1 += __shfl_xor(a1, off, 32);
        a2 += __shfl_xor(a2, off, 32);
    }
    if (lane == 0) { s1[n] = a1; s2[n] = a2; }
}

// ---------------------------------------------------------------------------
// Per-layer init: hnext = bias (segment_sum identity + b), mkey=-inf, denom=0
// ---------------------------------------------------------------------------
__global__ void init_layer_kernel(unsigned* __restrict__ mkey,
                                  float* __restrict__ denom,
                                  float* __restrict__ hnext,
                                  const float* __restrict__ bias) {
    const int n = blockIdx.x, c = threadIdx.x;
    if (c == 0) { mkey[n] = fkey(-INFINITY); denom[n] = 0.f; }
    hnext[(size_t)n * HID + c] = bias[c];
}

// ---------------------------------------------------------------------------
// alpha[e] = leaky_relu(s1[src]+s2[dst]+elog[e][layer]); atomic segment max.
// ---------------------------------------------------------------------------
__global__ void alpha_kernel(const long long* __restrict__ src,
                             const long long* __restrict__ dst,
                             const float* __restrict__ s1,
                             const float* __restrict__ s2,
                             const float* __restrict__ elog, int layer,
                             float* __restrict__ alpha,
                             unsigned* __restrict__ mkey, int E) {
    int e = blockIdx.x * blockDim.x + threadIdx.x;
    if (e >= E) return;
    int s = (int)src[e], d = (int)dst[e];
    float a = s1[s] + s2[d] + elog[(size_t)e * NLAY + layer];
    a = a > 0.f ? a : 0.2f * a;
    alpha[e] = a;
    atomicMax(&mkey[d], fkey(a));
}

// ---------------------------------------------------------------------------
// ea[e] = exp(alpha[e] - m[dst]); denom[dst] += ea[e]
// ---------------------------------------------------------------------------
__global__ void ea_kernel(const long long* __restrict__ dst,
                          const float* __restrict__ alpha,
                          const unsigned* __restrict__ mkey,
                          float* __restrict__ ea,
                          float* __restrict__ denom, int E) {
    int e = blockIdx.x * blockDim.x + threadIdx.x;
    if (e >= E) return;
    int d = (int)dst[e];
    float m = funkey(mkey[d]);
    if (!isfinite(m)) m = 0.f;
    float v = expf(alpha[e] - m);
    ea[e] = v;
    atomicAdd(&denom[d], v);
}

// ---------------------------------------------------------------------------
// hnext[dst] += hW[src] * (ea[e]/(denom[dst]+1e-16)); one block per edge.
// ---------------------------------------------------------------------------
__global__ void scatter_kernel(const long long* __restrict__ src,
                               const long long* __restrict__ dst,
                               const float* __restrict__ hW,
                               const float* __restrict__ ea,
                               const float* __restrict__ denom,
                               float* __restrict__ hnext) {
    const int e = blockIdx.x, c = threadIdx.x;
    const int s = (int)src[e], d = (int)dst[e];
    const float coef = ea[e] / (denom[d] + 1e-16f);
    atomicAdd(&hnext[(size_t)d * HID + c], hW[(size_t)s * HID + c] * coef);
}

// ---------------------------------------------------------------------------
// W3s[k][n] = W3[k][n] + W3[k+256][n]   (concat(h,h)@W3 == h@W3s)
// ---------------------------------------------------------------------------
__global__ void w3sum_kernel(const float* __restrict__ W3,
                             float* __restrict__ W3s) {
    int i = blockIdx.x * blockDim.x + threadIdx.x;
    if (i >= HID * HID) return;
    W3s[i] = W3[i] + W3[HID * HID + i];
}

// ---------------------------------------------------------------------------
extern "C" void kernel_launch(void* const* d_in, const int* in_sizes, int n_in,
                              void* d_out, int out_size, void* d_ws, size_t ws_size,
                              hipStream_t stream) {
    const float*     x    = (const float*)d_in[0];
    const long long* eidx = (const long long*)d_in[1];   // int64 [2, E]
    const float*     ea_t = (const float*)d_in[2];       // edge_attr
    const float*     W1   = (const float*)d_in[3];
    const float*     W2   = (const float*)d_in[4];
    const float*     Wc   = (const float*)d_in[5];       // [6,256,256]
    const float*     We   = (const float*)d_in[6];       // [6,768,256]
    const float*     aS   = (const float*)d_in[7];       // [6,256]
    const float*     aD   = (const float*)d_in[8];
    const float*     aE   = (const float*)d_in[9];
    const float*     bias = (const float*)d_in[10];      // [6,256]
    const float*     W3   = (const float*)d_in[11];      // [512,256]
    float*           out  = (float*)d_out;

    const long long* src = eidx;
    const long long* dst = eidx + NEDGES;

    // workspace carve-up (floats)
    float* ws = (float*)d_ws;
    size_t off = 0;
    float* hA    = ws + off; off += (size_t)NNODES * HID;
    float* hB    = ws + off; off += (size_t)NNODES * HID;
    float* hW    = ws + off; off += (size_t)NNODES * HID;
    float* elog  = ws + off; off += (size_t)NEDGES * NLAY;
    float* Vt    = ws + off; off += (size_t)NLAY * EDIM;
    float* s1    = ws + off; off += 50048;
    float* s2    = ws + off; off += 50048;
    unsigned* mk = (unsigned*)(ws + off); off += 50048;
    float* denom = ws + off; off += 50048;
    float* alpha = ws + off; off += NEDGES;
    float* eaB   = ws + off; off += NEDGES;
    float* W3s   = ws + off; off += (size_t)HID * HID;

    const dim3 gemmGrid((NNODES + 127) / 128, 4);
    const dim3 gemmBlk(256);

    // --- one-time prep ---
    ve_kernel<<<(NLAY * EDIM + 255) / 256, 256, 0, stream>>>(We, aE, Vt);
    edge_logits_kernel<<<(NEDGES + 7) / 8, 256, 0, stream>>>(ea_t, Vt, elog, NEDGES);
    w3sum_kernel<<<(HID * HID + 255) / 256, 256, 0, stream>>>(W3, W3s);

    // --- h = x @ W1 @ W2 ---
    gemm256_wmma_f32<0><<<gemmGrid, gemmBlk, 0, stream>>>(x, W1, hA, NNODES);
    gemm256_wmma_f32<0><<<gemmGrid, gemmBlk, 0, stream>>>(hA, W2, hB, NNODES);

    float* h = hB;
    float* hn = hA;
    for (int layer = 0; layer < NLAY; ++layer) {
        gemm256_wmma_f32<0><<<gemmGrid, gemmBlk, 0, stream>>>(
            h, Wc + (size_t)layer * HID * HID, hW, NNODES);
        node_scores_kernel<<<(NNODES + 7) / 8, 256, 0, stream>>>(
            hW, aS + (size_t)layer * HID, aD + (size_t)layer * HID, s1, s2, NNODES);
        init_layer_kernel<<<NNODES, HID, 0, stream>>>(
            mk, denom, hn, bias + (size_t)layer * HID);
        alpha_kernel<<<(NEDGES + 255) / 256, 256, 0, stream>>>(
            src, dst, s1, s2, elog, layer, alpha, mk, NEDGES);
        ea_kernel<<<(NEDGES + 255) / 256, 256, 0, stream>>>(
            dst, alpha, mk, eaB, denom, NEDGES);
        scatter_kernel<<<NEDGES, HID, 0, stream>>>(src, dst, hW, eaB, denom, hn);
        float* t = h; h = hn; hn = t;
    }

    // --- out = relu(concat(h,h)) @ W3  ==  relu(h) @ W3s ---
    gemm256_wmma_f32<1><<<gemmGrid, gemmBlk, 0, stream>>>(h, W3s, out, NNODES);
}